// DFConv_13615046329009
// MI455X (gfx1250) — compile-verified
//
#include <hip/hip_runtime.h>
#include <math.h>

typedef __attribute__((ext_vector_type(2))) float v2f;
typedef __attribute__((ext_vector_type(8))) float v8f;

#define Bn   8
#define Cc   64
#define Hh   128
#define Ww   128
#define HW   (Hh*Ww)
#define KK9  9
#define KDIM (Cc*KK9)   /* 576 */
#define COUT 64

/* workspace layout (in floats) */
#define WOT2_OFF 0                       /* 576*32 padded/transposed offset weights */
#define WT2_OFF  (576*32)                /* 576*64 transposed main weights          */
#define DY_OFF   (WT2_OFF + 576*64)
#define DX_OFF   (DY_OFF + Bn*KK9*HW)
#define MASK_OFF (DX_OFF + Bn*KK9*HW)

/* ---- kernel 0: reorganize weights into WMMA-friendly interleaved layout ----
 * Layout [k>>1][n][k&1]: a B-fragment (b.x=B[k][n], b.y=B[k+1][n]) is one
 * contiguous float2; consecutive lanes (n) read consecutive float2 -> coalesced.
 */
__global__ void prep_weights(const float* __restrict__ w_off,
                             const float* __restrict__ w_conv,
                             float* __restrict__ ws) {
    int i = blockIdx.x * blockDim.x + threadIdx.x;
    float* WOT2 = ws + WOT2_OFF;
    float* WT2  = ws + WT2_OFF;
    if (i < 576 * 64) {
        int k = i >> 6, o = i & 63;
        WT2[(k >> 1) * 128 + o * 2 + (k & 1)] = w_conv[o * 576 + k];
    }
    if (i < 576 * 32) {
        int k = i >> 5, n = i & 31;
        WOT2[(k >> 1) * 64 + n * 2 + (k & 1)] = (n < 27) ? w_off[n * 576 + k] : 0.f;
    }
}

/* ---- kernel 1: offset conv (3x3, 64->27) as im2col + WMMA f32 GEMM ----
 * One workgroup = 16 contiguous w positions at (b,h). A-tile in LDS,
 * interleaved [k>>1][m][k&1] so the A-fragment is one 8B LDS load/lane.
 */
__global__ __launch_bounds__(128) void offset_conv(
        const float* __restrict__ x, const float* __restrict__ bias,
        float* __restrict__ ws) {
    __shared__ float A2[KDIM * 16];
    const int wbase = blockIdx.x * 16, h = blockIdx.y, b = blockIdx.z;
    const int t = threadIdx.x;
    const int m = t & 15, g = t >> 4;

    const float* xb = x + (size_t)b * Cc * HW;
    for (int j = 0; j < 72; ++j) {              /* 8 groups x 72 = 576 (c,kp) */
        int p = g * 72 + j;
        int c = p / 9, kp = p - c * 9;
        int yy = h + kp / 3 - 1;
        int xx = wbase + m + (kp % 3) - 1;
        float v = 0.f;
        if (yy >= 0 && yy < Hh && xx >= 0 && xx < Ww) v = xb[c * HW + yy * Ww + xx];
        int k = c * 9 + kp;
        A2[(k >> 1) * 32 + m * 2 + (k & 1)] = v;
    }
    __syncthreads();

    const int wave = t >> 5;
    if (wave < 2) {                              /* N = 32 (27 padded) -> 2 waves */
        const int lane = t & 31;
        const int n = lane & 15, hi = lane >> 4; /* A row = n, K-pair sel = hi */
        const float* WOT2 = ws + WOT2_OFF;
        v8f acc = {};
        #pragma unroll 4
        for (int k0 = 0; k0 < KDIM; k0 += 4) {
            int kq = (k0 >> 1) + hi;
            v2f a  = *(const v2f*)&A2[kq * 32 + n * 2];
            v2f bb = *(const v2f*)&WOT2[kq * 64 + (wave * 16 + n) * 2];
            acc = __builtin_amdgcn_wmma_f32_16x16x4_f32(
                      false, a, false, bb, (short)0, acc, false, false);
        }
        float* dy  = ws + DY_OFF;
        float* dx  = ws + DX_OFF;
        float* msk = ws + MASK_OFF;
        int nch = wave * 16 + n;
        if (nch < 27) {
            float bv = bias[nch];
            for (int r = 0; r < 8; ++r) {
                int mm = r + 8 * hi;             /* C/D layout: M = r + 8*hi */
                int w  = wbase + mm;
                float v = acc[r] + bv;
                if (nch < 18) {                   /* offset channels: (kp,{dy,dx}) */
                    int kp = nch >> 1;
                    float* tgt = (nch & 1) ? dx : dy;
                    tgt[((b * KK9 + kp) * Hh + h) * Ww + w] = v;
                } else {                          /* mask channels: sigmoid */
                    int kp = nch - 18;
                    msk[((b * KK9 + kp) * Hh + h) * Ww + w] = 1.f / (1.f + expf(-v));
                }
            }
        }
    }
}

/* ---- kernel 2: bilinear sampling + main 16x64x576 WMMA GEMM ---- */
__global__ __launch_bounds__(128) void dfconv_main(
        const float* __restrict__ x, const float* __restrict__ ws,
        float* __restrict__ out) {
    __shared__ float A2[KDIM * 16];
    __shared__ int   sIdx[4][144];
    __shared__ float sW[4][144];
    const int wbase = blockIdx.x * 16, h = blockIdx.y, b = blockIdx.z;
    const int t = threadIdx.x;

    const float* dyp  = ws + DY_OFF;
    const float* dxp  = ws + DX_OFF;
    const float* mskp = ws + MASK_OFF;

    /* phase 0: bilinear corner idx/weights per (m,kp); mask folded into weights */
    for (int p = t; p < 144; p += 128) {
        int m = p & 15, kp = p >> 4;
        int w = wbase + m;
        int oidx = ((b * KK9 + kp) * Hh + h) * Ww + w;
        float ys = (float)(h - 1 + kp / 3) + dyp[oidx];
        float xs = (float)(w - 1 + kp % 3) + dxp[oidx];
        float mk = mskp[oidx];
        float y0f = floorf(ys), x0f = floorf(xs);
        float wy1 = ys - y0f, wx1 = xs - x0f;
        float wy0 = 1.f - wy1, wx0 = 1.f - wx1;
        int y0 = (int)y0f, x0 = (int)x0f;
        int   yc[2] = { y0, y0 + 1 }, xc[2] = { x0, x0 + 1 };
        float wy[2] = { wy0, wy1 },  wx[2] = { wx0, wx1 };
        for (int jy = 0; jy < 2; ++jy)
            for (int jx = 0; jx < 2; ++jx) {
                int j = jy * 2 + jx;
                int yy = yc[jy], xx = xc[jx];
                bool valid = (yy >= 0 && yy < Hh && xx >= 0 && xx < Ww);
                sIdx[j][p] = valid ? (yy * Ww + xx) : 0;
                sW[j][p]   = valid ? (wy[jy] * wx[jx] * mk) : 0.f;
            }
    }
    __syncthreads();

    /* phase 1: gather-sample all 64 channels into the LDS A-tile */
    {
        int m = t & 15, g = t >> 4;
        const float* xb = x + (size_t)b * Cc * HW;
        for (int j = 0; j < 72; ++j) {
            int pr = g * 72 + j;
            int c = pr / 9, kp = pr - c * 9;
            int p = kp * 16 + m;
            const float* xc = xb + c * HW;
            float v = sW[0][p] * xc[sIdx[0][p]] + sW[1][p] * xc[sIdx[1][p]]
                    + sW[2][p] * xc[sIdx[2][p]] + sW[3][p] * xc[sIdx[3][p]];
            int k = c * 9 + kp;
            A2[(k >> 1) * 32 + m * 2 + (k & 1)] = v;
        }
    }
    __syncthreads();

    /* phase 2: D[16x16] += A[16x576] * WT[576x16] per wave (4 waves cover N=64) */
    const int lane = t & 31, wave = t >> 5;
    const int n = lane & 15, hi = lane >> 4;
    const float* WT2 = ws + WT2_OFF;
    v8f acc = {};
    #pragma unroll 4
    for (int k0 = 0; k0 < KDIM; k0 += 4) {
        int kq = (k0 >> 1) + hi;
        v2f a  = *(const v2f*)&A2[kq * 32 + n * 2];
        v2f bb = *(const v2f*)&WT2[kq * 128 + (wave * 16 + n) * 2];
        acc = __builtin_amdgcn_wmma_f32_16x16x4_f32(
                  false, a, false, bb, (short)0, acc, false, false);
    }
    __syncthreads();                 /* all waves done reading A2 */

    /* phase 3: transpose D through LDS (reusing A2) for coalesced stores */
    float* Dw = A2 + wave * (16 * 17);
    for (int r = 0; r < 8; ++r)
        Dw[(r + 8 * hi) * 17 + n] = acc[r];
    __syncthreads();
    for (int r = 0; r < 8; ++r) {
        int o_local = r * 2 + hi;
        int mm = n;                  /* 16 consecutive lanes -> consecutive w */
        int o = wave * 16 + o_local;
        out[(((size_t)b * COUT + o) * Hh + h) * Ww + wbase + mm] = Dw[mm * 17 + o_local];
    }
}

extern "C" void kernel_launch(void* const* d_in, const int* in_sizes, int n_in,
                              void* d_out, int out_size, void* d_ws, size_t ws_size,
                              hipStream_t stream) {
    const float* x        = (const float*)d_in[0];
    const float* w_offset = (const float*)d_in[1];
    const float* b_offset = (const float*)d_in[2];
    const float* w_conv   = (const float*)d_in[3];
    float* out = (float*)d_out;
    float* ws  = (float*)d_ws;

    prep_weights<<<dim3((576 * 64 + 255) / 256), dim3(256), 0, stream>>>(
        w_offset, w_conv, ws);
    offset_conv<<<dim3(Ww / 16, Hh, Bn), dim3(128), 0, stream>>>(x, b_offset, ws);
    dfconv_main<<<dim3(Ww / 16, Hh, Bn), dim3(128), 0, stream>>>(x, ws, out);
}